// PhysicsGATLayer_70196945486385
// MI455X (gfx1250) — compile-verified
//
#include <hip/hip_runtime.h>
#include <hip/hip_bf16.h>

typedef __attribute__((ext_vector_type(2))) float v2f;
typedef __attribute__((ext_vector_type(8))) float v8f;
typedef int async_v4i __attribute__((vector_size(16)));   // b128 payload type

#define GAT_ALPHA 0.2f
#define GAT_NEG_INF -1000000000.0f

#if __has_builtin(__builtin_amdgcn_global_load_async_to_lds_b128) && \
    __has_builtin(__builtin_amdgcn_s_wait_asynccnt)
#define USE_ASYNC_LDS 1
#else
#define USE_ASYNC_LDS 0
#endif

// ---------------------------------------------------------------------------
// Kernel 1: h_prime = h @ W   ((B*N)x128 @ 128x64) with V_WMMA_F32_16X16X4_F32
// One wave (32 lanes) computes a 16x64 output tile (4 accumulators of 16x16).
// ---------------------------------------------------------------------------
__global__ __launch_bounds__(256) void k_hprime(const float* __restrict__ h,
                                                const float* __restrict__ W,
                                                float* __restrict__ hp) {
  const int wave = threadIdx.x >> 5;
  const int lane = threadIdx.x & 31;
  const int half = lane >> 4;      // 0: lanes 0-15, 1: lanes 16-31
  const int lm   = lane & 15;
  const int tile = blockIdx.x * 8 + wave;   // 1024 tiles of 16 rows (16384 rows)
  const int m0   = tile * 16;

  v8f acc0 = {}, acc1 = {}, acc2 = {}, acc3 = {};
  const float* __restrict__ arow = h + (size_t)(m0 + lm) * 128;

  for (int kk = 0; kk < 32; ++kk) {
    const int k0 = kk * 4 + half * 2;
    // A-frag (16x4 f32): lanes 0-15 hold K=0,1 ; lanes 16-31 hold K=2,3
    v2f af;
    af.x = arow[k0];
    af.y = arow[k0 + 1];
    // B-frags (4x16 f32): row-striped across lanes, VGPR0=K(half*2), VGPR1=K(half*2+1)
    const float* __restrict__ wr0 = W + (size_t)k0 * 64;
    const float* __restrict__ wr1 = wr0 + 64;
    v2f b0, b1, b2, b3;
    b0.x = wr0[lm];      b0.y = wr1[lm];
    b1.x = wr0[16 + lm]; b1.y = wr1[16 + lm];
    b2.x = wr0[32 + lm]; b2.y = wr1[32 + lm];
    b3.x = wr0[48 + lm]; b3.y = wr1[48 + lm];

    acc0 = __builtin_amdgcn_wmma_f32_16x16x4_f32(false, af, false, b0, (short)0, acc0, false, false);
    acc1 = __builtin_amdgcn_wmma_f32_16x16x4_f32(false, af, false, b1, (short)0, acc1, false, false);
    acc2 = __builtin_amdgcn_wmma_f32_16x16x4_f32(false, af, false, b2, (short)0, acc2, false, false);
    acc3 = __builtin_amdgcn_wmma_f32_16x16x4_f32(false, af, false, b3, (short)0, acc3, false, false);
  }

  // C/D layout: VGPR r holds M=r (lanes 0-15) / M=r+8 (lanes 16-31), N=lane%16
  for (int r = 0; r < 8; ++r) {
    const int row = m0 + r + half * 8;
    float* __restrict__ orow = hp + (size_t)row * 64;
    orow[lm]      = acc0[r];
    orow[16 + lm] = acc1[r];
    orow[32 + lm] = acc2[r];
    orow[48 + lm] = acc3[r];
  }
}

// ---------------------------------------------------------------------------
// Kernel 2: e_l = h_prime @ a[:64], e_r = h_prime @ a[64:]  (one thread / row)
// ---------------------------------------------------------------------------
__global__ __launch_bounds__(256) void k_el_er(const float* __restrict__ hp,
                                               const float* __restrict__ a,
                                               float* __restrict__ el,
                                               float* __restrict__ er) {
  const int row = blockIdx.x * 256 + threadIdx.x;   // 16384 rows
  const float* __restrict__ r = hp + (size_t)row * 64;
  float sl = 0.f, sr = 0.f;
#pragma unroll 8
  for (int c = 0; c < 64; ++c) {
    const float v = r[c];
    sl += v * a[c];
    sr += v * a[64 + c];
  }
  el[row] = sl;
  er[row] = sr;
}

// ---------------------------------------------------------------------------
// Kernel 3: row softmax of masked leaky_relu(e_l[i] + e_r[j]) -> attention row
// One 256-thread block per (b, i); 2048 columns, 8 per thread (coalesced).
// This is the bandwidth hog (134 MB streamed to HBM/L2) - keep it pure streaming.
// ---------------------------------------------------------------------------
__global__ __launch_bounds__(256) void k_softmax(const float* __restrict__ el,
                                                 const float* __restrict__ er,
                                                 const int* __restrict__ adj,
                                                 float* __restrict__ att) {
  const int i   = blockIdx.x;
  const int b   = blockIdx.y;
  const int tid = threadIdx.x;

  const float eli = el[b * 2048 + i];
  const float* __restrict__ erb  = er + b * 2048;
  const int* __restrict__   arow = adj + (size_t)i * 2048;

  float vals[8];
  float m = -3.4e38f;
#pragma unroll
  for (int k = 0; k < 8; ++k) {
    const int j = tid + k * 256;
    float e = eli + erb[j];
    e = (e > 0.f) ? e : GAT_ALPHA * e;          // leaky_relu
    e = (arow[j] > 0) ? e : GAT_NEG_INF;        // adjacency mask
    vals[k] = e;
    m = fmaxf(m, e);
  }

  __shared__ float red[256];
  red[tid] = m;
  __syncthreads();
  for (int s = 128; s > 0; s >>= 1) {
    if (tid < s) red[tid] = fmaxf(red[tid], red[tid + s]);
    __syncthreads();
  }
  const float rowmax = red[0];
  __syncthreads();

  float ssum = 0.f;
#pragma unroll
  for (int k = 0; k < 8; ++k) {
    vals[k] = __expf(vals[k] - rowmax);
    ssum += vals[k];
  }
  red[tid] = ssum;
  __syncthreads();
  for (int s = 128; s > 0; s >>= 1) {
    if (tid < s) red[tid] += red[tid + s];
    __syncthreads();
  }
  const float inv = 1.f / red[0];

  float* __restrict__ orow = att + ((size_t)b * 2048 + i) * 2048;
#pragma unroll
  for (int k = 0; k < 8; ++k) {
    const int j = tid + k * 256;
    orow[j] = vals[k] * inv;
  }
}

// ---------------------------------------------------------------------------
// Async staging of a 128x64 fp32 chunk (32 KB, contiguous) into LDS.
// 256 threads x 16 B x 8 passes. Tracked with ASYNCcnt on gfx1250.
// ---------------------------------------------------------------------------
__device__ __forceinline__ void stage_chunk(float* dst_lds,
                                            const float* __restrict__ gsrc,
                                            int tid) {
#pragma unroll
  for (int p = 0; p < 8; ++p) {
    const int off = (p * 256 + tid) * 4;   // float index, 16 bytes per thread
#if USE_ASYNC_LDS
    __builtin_amdgcn_global_load_async_to_lds_b128(
        (async_v4i*)(gsrc + off),
        (__attribute__((address_space(3))) async_v4i*)(dst_lds + off),
        0, 0);
#else
    *(float4*)(dst_lds + off) = *(const float4*)(gsrc + off);
#endif
  }
}

__device__ __forceinline__ void wait_async_all() {
#if USE_ASYNC_LDS
  __builtin_amdgcn_s_wait_asynccnt(0);
#endif
}

// ---------------------------------------------------------------------------
// Kernel 4: h_output = attention @ h_prime  (batched 2048x64x2048 fp32 WMMA)
// Each wave computes a 64x64 output tile (4 m-tiles x 4 n-tiles = 16 accums),
// so one set of B frags feeds 16 WMMAs. The shared h_prime K-chunk (128x64)
// is double-buffered in LDS via async global->LDS DMA, overlapped with math.
// ---------------------------------------------------------------------------
__global__ __launch_bounds__(256) void k_av(const float* __restrict__ att,
                                            const float* __restrict__ hp,
                                            float* __restrict__ out) {
  const int wave = threadIdx.x >> 5;
  const int lane = threadIdx.x & 31;
  const int half = lane >> 4;
  const int lm   = lane & 15;
  const int tid  = threadIdx.x;

  // 256 wave-tasks total: 8 batches x 32 tasks (64 rows each). 8 waves/block.
  const int gid = blockIdx.x * 8 + wave;
  const int b   = gid >> 5;            // all waves of a block share one batch
  const int m0  = (gid & 31) * 64;

  const float* __restrict__ A = att + (size_t)b * 2048 * 2048;
  const float* __restrict__ V = hp  + (size_t)b * 2048 * 64;

  __shared__ float lbuf[2][128 * 64];  // 2 x 32 KB K-chunks of h_prime[b]

  const float* __restrict__ arow0 = A + (size_t)(m0 +  0 + lm) * 2048;
  const float* __restrict__ arow1 = A + (size_t)(m0 + 16 + lm) * 2048;
  const float* __restrict__ arow2 = A + (size_t)(m0 + 32 + lm) * 2048;
  const float* __restrict__ arow3 = A + (size_t)(m0 + 48 + lm) * 2048;

  v8f acc[4][4];
#pragma unroll
  for (int mt = 0; mt < 4; ++mt)
#pragma unroll
    for (int nt = 0; nt < 4; ++nt) acc[mt][nt] = (v8f){};

  const int NCHUNK = 16;               // K = 2048 = 16 x 128
  stage_chunk(lbuf[0], V, tid);

  for (int c = 0; c < NCHUNK; ++c) {
    const int sel   = c & 1;
    const int kbase = c * 128;

    wait_async_all();                  // chunk c landed in lbuf[sel]
    __syncthreads();                   // visible to all waves; prior reads done

    if (c + 1 < NCHUNK)
      stage_chunk(lbuf[sel ^ 1], V + (size_t)(kbase + 128) * 64, tid);

    if (kbase + 128 < 2048) {          // prefetch next A stripe (global_prefetch_b8)
      __builtin_prefetch(arow0 + kbase + 128, 0, 0);
      __builtin_prefetch(arow1 + kbase + 128, 0, 0);
      __builtin_prefetch(arow2 + kbase + 128, 0, 0);
      __builtin_prefetch(arow3 + kbase + 128, 0, 0);
    }

    const float* lb = lbuf[sel];
#pragma unroll 2
    for (int kk = 0; kk < 32; ++kk) {
      const int k0l = kk * 4 + half * 2;     // K within chunk for this half-wave
      const int k0g = kbase + k0l;

      // B frags from LDS (shared by all 4 m-tiles)
      const float* __restrict__ l0 = lb + (size_t)k0l * 64;
      const float* __restrict__ l1 = l0 + 64;
      v2f bf[4];
#pragma unroll
      for (int nt = 0; nt < 4; ++nt) {
        bf[nt].x = l0[nt * 16 + lm];
        bf[nt].y = l1[nt * 16 + lm];
      }

      // A frags: one contiguous b64 load per m-tile
      v2f af[4];
      af[0].x = arow0[k0g]; af[0].y = arow0[k0g + 1];
      af[1].x = arow1[k0g]; af[1].y = arow1[k0g + 1];
      af[2].x = arow2[k0g]; af[2].y = arow2[k0g + 1];
      af[3].x = arow3[k0g]; af[3].y = arow3[k0g + 1];

#pragma unroll
      for (int mt = 0; mt < 4; ++mt)
#pragma unroll
        for (int nt = 0; nt < 4; ++nt)
          acc[mt][nt] = __builtin_amdgcn_wmma_f32_16x16x4_f32(
              false, af[mt], false, bf[nt], (short)0, acc[mt][nt], false, false);
    }
    __syncthreads();                   // all reads of lbuf[sel] done before reuse
  }

#pragma unroll
  for (int mt = 0; mt < 4; ++mt) {
#pragma unroll
    for (int r = 0; r < 8; ++r) {
      const int row = m0 + mt * 16 + r + half * 8;
      float* __restrict__ orow = out + ((size_t)b * 2048 + row) * 64;
      orow[lm]      = acc[mt][0][r];
      orow[16 + lm] = acc[mt][1][r];
      orow[32 + lm] = acc[mt][2][r];
      orow[48 + lm] = acc[mt][3][r];
    }
  }
}

// ---------------------------------------------------------------------------
// Launch: inputs are h(f32), adj_mask(i32), W(f32), a(f32).
// d_out = [h_output (8*2048*64) | attention (8*2048*2048)] fp32.
// ---------------------------------------------------------------------------
extern "C" void kernel_launch(void* const* d_in, const int* in_sizes, int n_in,
                              void* d_out, int out_size, void* d_ws, size_t ws_size,
                              hipStream_t stream) {
  const float* h   = (const float*)d_in[0];
  const int*   adj = (const int*)d_in[1];
  const float* W   = (const float*)d_in[2];
  const float* a   = (const float*)d_in[3];

  const size_t BN = 8 * 2048;               // 16384 rows
  float* hp = (float*)d_ws;                 // 16384*64 floats = 4 MB
  float* el = hp + BN * 64;                 // 16384 floats
  float* er = el + BN;                      // 16384 floats

  float* hout = (float*)d_out;              // 8*2048*64
  float* att  = hout + BN * 64;             // 8*2048*2048

  k_hprime<<<128, 256, 0, stream>>>(h, W, hp);
  k_el_er<<<64, 256, 0, stream>>>(hp, a, el, er);
  k_softmax<<<dim3(2048, 8), 256, 0, stream>>>(el, er, adj, att);
  k_av<<<32, 256, 0, stream>>>(att, hp, hout);
}